// CMValidatedGate_50680614093580
// MI455X (gfx1250) — compile-verified
//
#include <hip/hip_runtime.h>
#include <hip/hip_bf16.h>
#include <math.h>

// Problem constants (from reference): N=8192, A=512, D=512, NN=3
#define NROWS 8192
#define AANCH 512
#define DDIM  512

typedef __attribute__((ext_vector_type(2))) float v2f;
typedef __attribute__((ext_vector_type(8))) float v8f;

// ---------------------------------------------------------------------------
// K1: G = anchors @ anchors^T  (512x512, K=512) via V_WMMA_F32_16X16X4_F32.
// One wave (32 threads) per 16x16 tile. grid = (32, 32).
// A 16x4 f32 layout: lanes 0-15 hold M=lane, VGPR0=K0, VGPR1=K1;
//                    lanes 16-31 hold M=lane-16, VGPR0=K2, VGPR1=K3.
// B 4x16 mirrored over N. C: VGPR v -> M = v + 8*(lane>=16), N = lane&15.
// ---------------------------------------------------------------------------
__global__ __launch_bounds__(32) void gram_wmma_kernel(
    const float* __restrict__ anchors, float* __restrict__ G) {
  const int lane = threadIdx.x;
  const int half = lane >> 4;   // 0 or 1
  const int l    = lane & 15;
  const int m0 = blockIdx.y * 16;
  const int n0 = blockIdx.x * 16;

  const float* arow = anchors + (size_t)(m0 + l) * DDIM + 2 * half;
  const float* brow = anchors + (size_t)(n0 + l) * DDIM + 2 * half;

  v8f c = {};
#if __has_builtin(__builtin_amdgcn_wmma_f32_16x16x4_f32)
  #pragma unroll 8
  for (int kb = 0; kb < DDIM; kb += 4) {
    v2f a = *(const v2f*)(arow + kb);   // A[m0+l][kb+2h .. +1]
    v2f b = *(const v2f*)(brow + kb);   // B[kb+2h..][n0+l] = anchors[n0+l][kb+2h..]
    c = __builtin_amdgcn_wmma_f32_16x16x4_f32(
        /*neg_a=*/false, a, /*neg_b=*/false, b,
        /*c_mod=*/(short)0, c, /*reuse_a=*/false, /*reuse_b=*/false);
  }
  float* gout = G + (size_t)(m0 + 8 * half) * AANCH + n0 + l;
  #pragma unroll
  for (int v = 0; v < 8; ++v) gout[(size_t)v * AANCH] = c[v];
#else
  // Scalar fallback (keeps this file compiling if the builtin is absent):
  // lane computes one output element per row chunk.
  for (int r = 0; r < 8; ++r) {
    int m = m0 + r + 8 * half;
    int n = n0 + l;
    float acc = 0.0f;
    for (int k = 0; k < DDIM; ++k)
      acc = fmaf(anchors[(size_t)m * DDIM + k], anchors[(size_t)n * DDIM + k], acc);
    G[(size_t)m * AANCH + n] = acc;
  }
#endif
}

// ---------------------------------------------------------------------------
// K2: per-anchor 3-NN (stable tiebreak: smaller index wins on equal dist)
// + 5x5 Cayley-Menger determinant via partial-pivot LU.
// One thread per anchor; single block of 512 threads.
// ---------------------------------------------------------------------------
__global__ __launch_bounds__(AANCH) void cm_kernel(
    const float* __restrict__ G, float* __restrict__ cm) {
  __shared__ float diag[AANCH];
  const int i = threadIdx.x;
  diag[i] = G[(size_t)i * AANCH + i];
  __syncthreads();

  const float di = diag[i];
  float bv0 = 3.0e38f, bv1 = 3.0e38f, bv2 = 3.0e38f;
  int   bi0 = 0, bi1 = 0, bi2 = 0;
  for (int j = 0; j < AANCH; ++j) {
    float g  = G[(size_t)j * AANCH + i];        // G symmetric -> coalesced reads
    float d2 = fmaxf(di + diag[j] - 2.0f * g, 0.0f);
    float d  = sqrtf(d2) + (j == i ? 1.0e12f : 0.0f);
    if (d < bv0)      { bv2=bv1; bi2=bi1; bv1=bv0; bi1=bi0; bv0=d; bi0=j; }
    else if (d < bv1) { bv2=bv1; bi2=bi1; bv1=d;  bi1=j; }
    else if (d < bv2) { bv2=d;  bi2=j; }
  }

  int idx[4] = { i, bi0, bi1, bi2 };
  float M[5][5];
  M[0][0] = 0.0f;
  #pragma unroll
  for (int c = 1; c < 5; ++c) { M[0][c] = 1.0f; M[c][0] = 1.0f; }
  for (int p = 0; p < 4; ++p)
    for (int q = 0; q < 4; ++q) {
      int gp = idx[p], gq = idx[q];
      float g = G[(size_t)gp * AANCH + gq];
      M[p + 1][q + 1] = diag[gp] + diag[gq] - 2.0f * g;  // no clamp (matches ref)
    }

  // det(5x5), partial pivoting
  float det = 1.0f;
  for (int c = 0; c < 5; ++c) {
    int piv = c; float pv = fabsf(M[c][c]);
    for (int r = c + 1; r < 5; ++r) {
      float v = fabsf(M[r][c]);
      if (v > pv) { pv = v; piv = r; }
    }
    if (piv != c) {
      for (int k = 0; k < 5; ++k) { float t = M[c][k]; M[c][k] = M[piv][k]; M[piv][k] = t; }
      det = -det;
    }
    float p = M[c][c];
    det *= p;
    if (p != 0.0f) {
      float inv = 1.0f / p;
      for (int r = c + 1; r < 5; ++r) {
        float f = M[r][c] * inv;
        for (int k = c + 1; k < 5; ++k) M[r][k] = fmaf(-f, M[c][k], M[r][k]);
      }
    }
  }
  // sign_corr = (-1)^K with K=4 -> +1
  float s = (det > 0.0f) ? 1.0f : ((det < 0.0f) ? -1.0f : 0.0f);
  cm[i] = s * logf(fabsf(det) + 1e-12f);
}

// ---------------------------------------------------------------------------
// K3: normalize anchor_cm (mean, std with ddof=1, clamp 1e-8) and precompute
// per-anchor hidden bias table preA[a][h] = W1[h][0]*cm_norm[a] + b1[h].
// ---------------------------------------------------------------------------
__global__ __launch_bounds__(AANCH) void norm_prep_kernel(
    const float* __restrict__ cm, const float* __restrict__ W1,
    const float* __restrict__ b1, float* __restrict__ preA) {
  __shared__ float red[AANCH];
  __shared__ float mean_s, std_s;
  const int t = threadIdx.x;
  const float x = cm[t];

  red[t] = x; __syncthreads();
  for (int s = AANCH / 2; s > 0; s >>= 1) {
    if (t < s) red[t] += red[t + s];
    __syncthreads();
  }
  if (t == 0) mean_s = red[0] * (1.0f / AANCH);
  __syncthreads();
  const float mean = mean_s;
  const float dlt = x - mean;
  red[t] = dlt * dlt; __syncthreads();
  for (int s = AANCH / 2; s > 0; s >>= 1) {
    if (t < s) red[t] += red[t + s];
    __syncthreads();
  }
  if (t == 0) std_s = fmaxf(sqrtf(red[0] * (1.0f / (AANCH - 1))), 1e-8f);
  __syncthreads();

  const float nrm = (x - mean) / std_s;
  #pragma unroll
  for (int h = 0; h < 16; ++h)
    preA[t * 16 + h] = fmaf(W1[h * 3 + 0], nrm, b1[h]);
}

// ---------------------------------------------------------------------------
// K4: per (row, anchor) element: stable rank via LDS-broadcast counting,
// then 3->16->1 gate MLP with exact-erf GELU and sigmoid.
// One block (512 threads) per row; 8192 blocks.
// ---------------------------------------------------------------------------
__global__ __launch_bounds__(AANCH) void gate_kernel(
    const float* __restrict__ tri, const float* __restrict__ preA,
    const float* __restrict__ W1, const float* __restrict__ W2,
    const float* __restrict__ b2, float* __restrict__ out) {
  __shared__ __align__(16) float srow[AANCH];
  const int n = blockIdx.x;
  const int j = threadIdx.x;
  const float my = tri[(size_t)n * AANCH + j];
  srow[j] = my;

  float w1c1[16], w1c2[16], w2v[16], pa[16];
  #pragma unroll
  for (int h = 0; h < 16; ++h) {
    w1c1[h] = W1[h * 3 + 1];
    w1c2[h] = W1[h * 3 + 2];
    w2v[h]  = W2[h];
  }
  #pragma unroll
  for (int h = 0; h < 16; ++h) pa[h] = preA[j * 16 + h];
  const float b2v = b2[0];
  __syncthreads();

  // stable rank = #{k: v<my} + #{k<j: v==my}  (matches double stable argsort)
  int cnt = 0;
  const float4* s4 = (const float4*)srow;
  #pragma unroll 4
  for (int k4 = 0; k4 < AANCH / 4; ++k4) {
    float4 v = s4[k4];            // all lanes same address -> LDS broadcast
    int kb = k4 << 2;
    cnt += (v.x < my) + (v.y < my) + (v.z < my) + (v.w < my);
    cnt += (v.x == my && kb     < j);
    cnt += (v.y == my && kb + 1 < j);
    cnt += (v.z == my && kb + 2 < j);
    cnt += (v.w == my && kb + 3 < j);
  }
  const float rank = (float)cnt * (1.0f / (AANCH - 1));
  const float cs = 1.0f - my;

  float logit = b2v;
  #pragma unroll
  for (int h = 0; h < 16; ++h) {
    float xh = fmaf(w1c1[h], cs, fmaf(w1c2[h], rank, pa[h]));
    float g  = 0.5f * xh * (1.0f + erff(xh * 0.70710678118654752f)); // exact GELU
    logit = fmaf(w2v[h], g, logit);
  }
  out[(size_t)n * AANCH + j] = 1.0f / (1.0f + expf(-logit));
}

// ---------------------------------------------------------------------------
// Launch: inputs (all f32) in setup_inputs() order:
//   0: embedding (UNUSED by reference) 1: anchors 2: tri 3: W1 4: b1 5: W2 6: b2
// Workspace layout: G (1 MiB) | cm (2 KiB, at +1 MiB) | preA (32 KiB, at +1 MiB+4 KiB)
// ---------------------------------------------------------------------------
extern "C" void kernel_launch(void* const* d_in, const int* in_sizes, int n_in,
                              void* d_out, int out_size, void* d_ws, size_t ws_size,
                              hipStream_t stream) {
  (void)in_sizes; (void)n_in; (void)out_size; (void)ws_size;
  const float* anchors = (const float*)d_in[1];
  const float* tri     = (const float*)d_in[2];
  const float* W1      = (const float*)d_in[3];
  const float* b1      = (const float*)d_in[4];
  const float* W2      = (const float*)d_in[5];
  const float* b2      = (const float*)d_in[6];
  float* out = (float*)d_out;

  char* ws   = (char*)d_ws;
  float* G    = (float*)ws;                                  // 512*512*4 = 1 MiB
  float* cm   = (float*)(ws + (1u << 20));                   // 2 KiB
  float* preA = (float*)(ws + (1u << 20) + 4096);            // 32 KiB

  gram_wmma_kernel<<<dim3(AANCH / 16, AANCH / 16), 32, 0, stream>>>(anchors, G);
  cm_kernel<<<1, AANCH, 0, stream>>>(G, cm);
  norm_prep_kernel<<<1, AANCH, 0, stream>>>(cm, W1, b1, preA);
  gate_kernel<<<NROWS, AANCH, 0, stream>>>(tri, preA, W1, W2, b2, out);
}